// SingleHeadAttention_8486855377372
// MI455X (gfx1250) — compile-verified
//
#include <hip/hip_runtime.h>

#define BB 64
#define SS 8192
#define IN_DIM 512
#define OUT_DIM 128
#define DVV 128
#define G 32                 // chunks per batch
#define CHUNK (SS / G)       // 256 s-values per block
#define THREADS 256          // 8 wave32
#define S_PER_WAVE (CHUNK / (THREADS / 32))   // 32

typedef float v2f __attribute__((ext_vector_type(2)));
typedef float v8f __attribute__((ext_vector_type(8)));

// workspace layout (floats): r[B][128], then partials[(b*G+g)] = {m, l, ctx[128]}
#define R_FLOATS (BB * OUT_DIM)
#define PART_STRIDE (2 + DVV)

// ---------------------------------------------------------------------------
// Kernel 1: r[b] = Wk . (q[b] . Wq) * (1/sqrt(OUT_DIM))   [tiny]
// ---------------------------------------------------------------------------
__global__ __launch_bounds__(128)
void proj_kernel(const float* __restrict__ q, const float* __restrict__ Wq,
                 const float* __restrict__ Wk, float* __restrict__ r) {
    __shared__ float qv[OUT_DIM];
    const int b = blockIdx.x;
    const int t = threadIdx.x;

    float s = 0.f;
    for (int d = 0; d < IN_DIM; ++d)
        s = fmaf(q[b * IN_DIM + d], Wq[d * OUT_DIM + t], s);
    qv[t] = s;
    __syncthreads();

    float s2 = 0.f;
    for (int e = 0; e < OUT_DIM; ++e)
        s2 = fmaf(Wk[t * OUT_DIM + e], qv[e], s2);
    r[b * OUT_DIM + t] = s2 * 0.08838834764831845f;   // 1/sqrt(128)
}

// ---------------------------------------------------------------------------
// Kernel 2: per (b, chunk): WMMA scores, local softmax stats, weighted v sum
// ---------------------------------------------------------------------------
__global__ __launch_bounds__(THREADS)
void attn_chunk_kernel(const float* __restrict__ k, const float* __restrict__ v,
                       const float* __restrict__ r, float* __restrict__ part) {
    __shared__ float r_lds[OUT_DIM];
    __shared__ float w_lds[CHUNK];
    __shared__ float red[THREADS];

    const int b    = blockIdx.x / G;
    const int g    = blockIdx.x % G;
    const int t    = threadIdx.x;
    const int lane = t & 31;
    const int wave = t >> 5;
    const int s0   = g * CHUNK;

    if (t < OUT_DIM) r_lds[t] = r[b * OUT_DIM + t];

    // CDNA5 prefetch of the v chunk we will need after the score phase
    __builtin_prefetch(v + ((size_t)b * SS + s0 + t) * DVV, 0, 1);
    __syncthreads();

    const int half = lane >> 4;    // 0: K pair {0,1}, 1: K pair {2,3}
    const int m16  = lane & 15;    // M row within tile

    // ---- scores via V_WMMA_F32_16X16X4_F32 (B columns all = r => D cols equal)
    for (int tile = 0; tile < S_PER_WAVE / 16; ++tile) {
        const int srow = s0 + wave * S_PER_WAVE + tile * 16;
        const float* krow = k + ((size_t)b * SS + srow + m16) * OUT_DIM;
        v8f c = {};
#pragma unroll 8
        for (int kk = 0; kk < OUT_DIM / 4; ++kk) {
            const int col = kk * 4 + half * 2;
            v2f a  = *(const v2f*)(krow + col);
            v2f bb = *(const v2f*)(&r_lds[col]);
            c = __builtin_amdgcn_wmma_f32_16x16x4_f32(
                    false, a, false, bb, (short)0, c, false, false);
        }
        // C layout: lane 0 holds M=0..7 in c[0..7]; lane 16 holds M=8..15
        if (m16 == 0) {
#pragma unroll
            for (int i = 0; i < 8; ++i)
                w_lds[wave * S_PER_WAVE + tile * 16 + half * 8 + i] = c[i];
        }
    }
    __syncthreads();

    // ---- block max over CHUNK == THREADS scores
    float sc = w_lds[t];
    red[t] = sc;
    __syncthreads();
    for (int o = THREADS / 2; o > 0; o >>= 1) {
        if (t < o) red[t] = fmaxf(red[t], red[t + o]);
        __syncthreads();
    }
    const float m = red[0];
    __syncthreads();

    // ---- exp and local sum
    const float e = __expf(sc - m);
    w_lds[t] = e;
    red[t]   = e;
    __syncthreads();
    for (int o = THREADS / 2; o > 0; o >>= 1) {
        if (t < o) red[t] += red[t + o];
        __syncthreads();
    }
    const float l = red[0];
    __syncthreads();

    // ---- ctx[j] = sum_s e_s * v[b, s0+s, j]  (coalesced 512B row loads)
    const int j = t & (DVV - 1);
    const int h = t >> 7;                      // split s-range in two halves
    const float* vbase = v + ((size_t)b * SS + s0 + h * (CHUNK / 2)) * DVV + j;
    float acc = 0.f;
#pragma unroll 4
    for (int i = 0; i < CHUNK / 2; ++i)
        acc = fmaf(w_lds[h * (CHUNK / 2) + i], vbase[(size_t)i * DVV], acc);
    red[t] = acc;
    __syncthreads();

    float* pb = part + (size_t)(b * G + g) * PART_STRIDE;
    if (t < DVV) pb[2 + t] = red[t] + red[t + 128];
    if (t == 0) { pb[0] = m; pb[1] = l; }
}

// ---------------------------------------------------------------------------
// Kernel 3: combine chunk partials (deterministic, no atomics)
// ---------------------------------------------------------------------------
__global__ __launch_bounds__(128)
void attn_combine_kernel(const float* __restrict__ part, float* __restrict__ out) {
    const int b = blockIdx.x;
    const int t = threadIdx.x;
    const float* pb = part + (size_t)b * G * PART_STRIDE;

    float M = -3.402823466e+38f;
    for (int g = 0; g < G; ++g) M = fmaxf(M, pb[g * PART_STRIDE + 0]);

    float L = 0.f, acc = 0.f;
    for (int g = 0; g < G; ++g) {
        const float sc = __expf(pb[g * PART_STRIDE + 0] - M);
        L   = fmaf(pb[g * PART_STRIDE + 1], sc, L);
        acc = fmaf(pb[g * PART_STRIDE + 2 + t], sc, acc);
    }
    out[b * DVV + t] = acc / L;
}

// ---------------------------------------------------------------------------
extern "C" void kernel_launch(void* const* d_in, const int* in_sizes, int n_in,
                              void* d_out, int out_size, void* d_ws, size_t ws_size,
                              hipStream_t stream) {
    const float* k  = (const float*)d_in[0];
    const float* q  = (const float*)d_in[1];
    const float* v  = (const float*)d_in[2];
    const float* Wk = (const float*)d_in[3];
    const float* Wq = (const float*)d_in[4];

    float* ws   = (float*)d_ws;
    float* r    = ws;                 // B*128 floats
    float* part = ws + R_FLOATS;      // B*G*130 floats (~1.04 MB total)

    proj_kernel<<<BB, 128, 0, stream>>>(q, Wq, Wk, r);
    attn_chunk_kernel<<<BB * G, THREADS, 0, stream>>>(k, v, r, part);
    attn_combine_kernel<<<BB, DVV, 0, stream>>>(part, (float*)d_out);
}